// Ranker_26482768347854
// MI455X (gfx1250) — compile-verified
//
#include <hip/hip_runtime.h>

typedef float v2f __attribute__((ext_vector_type(2)));
typedef float v4f __attribute__((ext_vector_type(4)));
typedef float v8f __attribute__((ext_vector_type(8)));

#define C_DIM 1024
#define P_DIM 4
#define ROWS_PER_WG 128
#define ROWS_PER_WAVE 16
#define KC 64
#define NCHUNK (C_DIM / KC)      // 16
#define XS_STRIDE 68             // 64 + 4 pad: 16B-aligned rows, 4-bank skew
#define WS_STRIDE 1028           // 1024 + 4 pad: 16B-aligned rows, 4-bank skew
#define WS_ROWS 5                // rows 0..3 = W, row 4 = zeros (shared N-pad row, broadcast reads)

__global__ __launch_bounds__(256) void ranker_wmma_f32_kernel(
    const float* __restrict__ x,
    const float* __restrict__ W,
    const float* __restrict__ bias,
    const int*   __restrict__ actions,
    const int*   __restrict__ is_global,
    float*       __restrict__ out)
{
    __shared__ float xs[2][ROWS_PER_WG * XS_STRIDE]; // 2 x 34816 B: double-buffered x tiles
    __shared__ float ws[WS_ROWS * WS_STRIDE];        // 20560 B: W rows 0-3 + one zero row

    const int tid  = threadIdx.x;
    const int lane = tid & 31;
    const int wave = tid >> 5;        // 0..7
    const int half = lane >> 4;       // 0/1 (A-frag K half, D-tile row half)
    const int l16  = lane & 15;
    const int wg_row0 = blockIdx.x * ROWS_PER_WG;

    // ---- Hoist tiny epilogue loads: they complete during the main loop ----
    const int   grow  = wg_row0 + wave * ROWS_PER_WAVE + l16;
    const int   act   = actions[grow] & (P_DIM - 1);
    const int   gflag = is_global[0];
    const float bs0 = bias[0], bs1 = bias[1], bs2 = bias[2], bs3 = bias[3];

    // ---- One-time: W[4][1024] -> LDS rows 0..3; zero row 4 ----
    for (int i = tid; i < (P_DIM * C_DIM) / 4; i += 256) {
        int r  = i >> 8;              // 256 float4 per row
        int c4 = i & 255;
        v4f v = *(const v4f*)(W + r * C_DIM + c4 * 4);
        *(v4f*)(ws + r * WS_STRIDE + c4 * 4) = v;
    }
    {
        const v4f z = {0.f, 0.f, 0.f, 0.f};
        float* zrow = ws + P_DIM * WS_STRIDE;         // 16B aligned
        for (int i = tid; i < WS_STRIDE / 4; i += 256)
            *(v4f*)(zrow + i * 4) = z;
    }

    // ---- Staging geometry: this thread owns 8 (row, col4) slots per chunk ----
    int srow[8], scol[8];
    #pragma unroll
    for (int p = 0; p < 8; ++p) {
        int idx = p * 256 + tid;
        srow[p] = idx >> 4;           // 0..127
        scol[p] = (idx & 15) * 4;     // 0..60
    }

    // ---- Prologue: chunk 0 -> regs -> xs[0] (x streamed once, NT: don't thrash L2) ----
    v4f pref[8];
    #pragma unroll
    for (int p = 0; p < 8; ++p)
        pref[p] = __builtin_nontemporal_load(
            (const v4f*)(x + (size_t)(wg_row0 + srow[p]) * C_DIM + scol[p]));
    #pragma unroll
    for (int p = 0; p < 8; ++p)
        *(v4f*)(&xs[0][0] + srow[p] * XS_STRIDE + scol[p]) = pref[p];

    v8f acc0 = {0.f, 0.f, 0.f, 0.f, 0.f, 0.f, 0.f, 0.f};
    v8f acc1 = {0.f, 0.f, 0.f, 0.f, 0.f, 0.f, 0.f, 0.f};

    const int lrow = wave * ROWS_PER_WAVE + l16;
    const float* arowA = &xs[0][0] + lrow * XS_STRIDE + 2 * half;
    const float* arowB = &xs[1][0] + lrow * XS_STRIDE + 2 * half;
    const float* brow  = ws + ((l16 < P_DIM) ? l16 : P_DIM) * WS_STRIDE + 2 * half;

    #pragma unroll 1
    for (int c = 0; c < NCHUNK; ++c) {
        __syncthreads();              // buf[c&1] readable; buf[(c+1)&1] writable

        // Issue next chunk's HBM loads FIRST: they fly during the WMMA phase,
        // and their loadcnt wait lands just before the tail LDS stores.
        if (c + 1 < NCHUNK) {
            #pragma unroll
            for (int p = 0; p < 8; ++p)
                pref[p] = __builtin_nontemporal_load(
                    (const v4f*)(x + (size_t)(wg_row0 + srow[p]) * C_DIM
                                   + (c + 1) * KC + scol[p]));
        }

        // ---- 16 WMMA steps (K=4), two independent accumulator chains ----
        const float* arow  = (c & 1) ? arowB : arowA;
        const int    kbase = c * KC;
        #pragma unroll
        for (int kk = 0; kk < KC; kk += 8) {
            v2f a0 = *(const v2f*)(arow + kk);
            v2f b0 = *(const v2f*)(brow + kbase + kk);
            v2f a1 = *(const v2f*)(arow + kk + 4);
            v2f b1 = *(const v2f*)(brow + kbase + kk + 4);
            acc0 = __builtin_amdgcn_wmma_f32_16x16x4_f32(
                false, a0, false, b0, (short)0, acc0, false, false);
            acc1 = __builtin_amdgcn_wmma_f32_16x16x4_f32(
                false, a1, false, b1, (short)0, acc1, false, false);
        }

        // ---- Tail: park next chunk in the other buffer ----
        if (c + 1 < NCHUNK) {
            float* xd = &xs[(c + 1) & 1][0];
            #pragma unroll
            for (int p = 0; p < 8; ++p)
                *(v4f*)(xd + srow[p] * XS_STRIDE + scol[p]) = pref[p];
        }
    }

    // ---- Scatter D-tile (16x16, stride 17) into xs[0] (free), gather selected head ----
    float* dt = &xs[0][0] + wave * (16 * 17);
    #pragma unroll
    for (int r = 0; r < 8; ++r) {
        int m = r + 8 * half;         // VGPR r holds rows r (lanes 0-15) and r+8 (lanes 16-31)
        dt[m * 17 + l16] = acc0[r] + acc1[r];
    }
    __syncthreads();

    if (half == 0) {                  // lanes 0-15: one output row each
        float result;
        if (gflag) {
            float s0 = dt[l16 * 17 + 0] + bs0;
            float s1 = dt[l16 * 17 + 1] + bs1;
            float s2 = dt[l16 * 17 + 2] + bs2;
            float s3 = dt[l16 * 17 + 3] + bs3;
            result = fmaxf(fmaxf(s0, s1), fmaxf(s2, s3));
        } else {
            float bsel = (act == 0) ? bs0 : (act == 1) ? bs1 : (act == 2) ? bs2 : bs3;
            result = dt[l16 * 17 + act] + bsel;
        }
        __builtin_nontemporal_store(result, out + grow);
    }
}

extern "C" void kernel_launch(void* const* d_in, const int* in_sizes, int n_in,
                              void* d_out, int out_size, void* d_ws, size_t ws_size,
                              hipStream_t stream) {
    const float* x         = (const float*)d_in[0];
    const float* W         = (const float*)d_in[1];
    const float* bias      = (const float*)d_in[2];
    const int*   actions   = (const int*)d_in[3];
    const int*   is_global = (const int*)d_in[4];
    float*       out       = (float*)d_out;

    const int Brows = in_sizes[0] / C_DIM;          // 65536
    dim3 grid(Brows / ROWS_PER_WG);                 // 512
    dim3 block(256);
    hipLaunchKernelGGL(ranker_wmma_f32_kernel, grid, block, 0, stream,
                       x, W, bias, actions, is_global, out);
}